// QuantizedEncoderFromSpecGenericNorm_50302656971356
// MI455X (gfx1250) — compile-verified
//
#include <hip/hip_runtime.h>
#include <hip/hip_bf16.h>
#include <stdint.h>

typedef __attribute__((ext_vector_type(2))) float v2f;
typedef __attribute__((ext_vector_type(4))) float v4f;
typedef __attribute__((ext_vector_type(8))) float v8f;
typedef __attribute__((ext_vector_type(4))) int   v4i;

#define LEAK 0.2f
#define EPSN 1e-5f
#define BATCH 8192
#define BM 128
#define BN 64
#define BK 16
#define LDP 20   // padded LDS row stride (floats): 80B = 16B-aligned rows, conflict-spread banks

#if defined(__has_builtin)
#if __has_builtin(__builtin_amdgcn_global_load_async_to_lds_b128)
#define HAS_ASYNC 1
#endif
#endif

__device__ __forceinline__ v8f wmma4(v2f a, v2f b, v8f c) {
  // D = A(16x4,f32) * B(4x16,f32) + C(16x16,f32)  -> v_wmma_f32_16x16x4_f32
  return __builtin_amdgcn_wmma_f32_16x16x4_f32(false, a, false, b, (short)0, c, false, false);
}

// 16-byte global -> LDS copy: async DMA (ASYNCcnt) when available, else sync v4f.
__device__ __forceinline__ void copy16(const float* g, float* l) {
#ifdef HAS_ASYNC
  __builtin_amdgcn_global_load_async_to_lds_b128(
      (__attribute__((address_space(1))) v4i*)(uintptr_t)g,
      (__attribute__((address_space(3))) v4i*)(uint32_t)(uintptr_t)l,
      0, 0);
#else
  *(v4f*)l = *(const v4f*)g;
#endif
}

__device__ __forceinline__ void async_wait_3() {
#ifdef HAS_ASYNC
#if __has_builtin(__builtin_amdgcn_s_wait_asynccnt)
  __builtin_amdgcn_s_wait_asynccnt(3);
#else
  asm volatile("s_wait_asynccnt 0x3" ::: "memory");
#endif
#endif
}
__device__ __forceinline__ void async_wait_0() {
#ifdef HAS_ASYNC
#if __has_builtin(__builtin_amdgcn_s_wait_asynccnt)
  __builtin_amdgcn_s_wait_asynccnt(0);
#else
  asm volatile("s_wait_asynccnt 0x0" ::: "memory");
#endif
#endif
}

// ---------------------------------------------------------------------------
// Generic grouped GEMM:  C[b, g*N + n] (+)= sum_k A'[b, g*K + k] * W[g, n, k]
// A' = NORM ? lrelu(A*scale+shift) : A (fused at fragment load).
// Double-buffered global->LDS pipeline (async DMA when the toolchain has it).
// ---------------------------------------------------------------------------
template<bool NORM, bool BIAS, bool RESID>
__global__ __launch_bounds__(256)
void gemm_wmma_f32(const float* __restrict__ A, const float* __restrict__ W,
                   float* __restrict__ Cout,
                   const float* __restrict__ scale, const float* __restrict__ shift,
                   const float* __restrict__ bias,
                   int M, int N, int K, int ldA, int ldC)
{
  __shared__ __align__(16) float lsA[2][BM][LDP];  // activations, row-major per stage
  __shared__ __align__(16) float lsB[2][BN][LDP];  // weights as W[n][k] (no transpose)
  __shared__ __align__(16) float lsS[2][16];       // bn scale slice
  __shared__ __align__(16) float lsH[2][16];       // bn shift slice
  (void)M;

  const int g      = blockIdx.y;
  const int ntiles = (N + BN - 1) / BN;
  const int tm     = blockIdx.x / ntiles;
  const int tn     = blockIdx.x % ntiles;
  const int tid    = threadIdx.x;
  const int lane   = tid & 31;
  const int wave   = tid >> 5;
  const int wm     = wave & 3;     // 4 waves tile M (4*32 = 128)
  const int wn     = wave >> 2;    // 2 waves tile N (2*32 = 64)
  const int l15    = lane & 15;
  const int hi     = lane >> 4;

  const int    gm0   = tm * BM;
  const int    gn0   = tn * BN;
  const size_t colA0 = (size_t)g * K;
  const size_t colC0 = (size_t)g * N;

  v8f acc00 = {}; v8f acc01 = {}; v8f acc10 = {}; v8f acc11 = {};

  // staging map: each thread moves 16B chunks; rows 16B-aligned in LDS (LDP*4 = 80)
  const int srow = tid >> 2;          // 0..63
  const int skq  = (tid & 3) << 2;    // 0,4,8,12

  auto issue_stage = [&](int buf, int k0) {
    // A tile: 128 rows x 16 k  (two 16B chunks per thread)
    copy16(&A[(size_t)(gm0 + srow)      * ldA + colA0 + k0 + skq], &lsA[buf][srow][skq]);
    copy16(&A[(size_t)(gm0 + srow + 64) * ldA + colA0 + k0 + skq], &lsA[buf][srow + 64][skq]);
    // B tile: 64 rows x 16 k, clamped so every wave issues exactly one op
    int gn = gn0 + srow; if (gn > N - 1) gn = N - 1;   // OOB cols discarded in epilogue
    copy16(&W[((size_t)g * N + gn) * K + k0 + skq], &lsB[buf][srow][skq]);
    if (NORM && tid < 16) {
      lsS[buf][tid] = scale[colA0 + k0 + tid];
      lsH[buf][tid] = shift[colA0 + k0 + tid];
    }
  };

  const int r0 = wm * 32 + l15;
  const int c0 = wn * 32 + l15;

  auto compute_stage = [&](int buf) {
    #pragma unroll
    for (int kk = 0; kk < BK; kk += 4) {
      const int ko = kk + 2 * hi;   // even -> 8B-aligned v2f (ds_load_b64)
      v2f a0 = *(const v2f*)&lsA[buf][r0][ko];
      v2f a1 = *(const v2f*)&lsA[buf][r0 + 16][ko];
      if (NORM) {
        v2f sc = *(const v2f*)&lsS[buf][ko];
        v2f sh = *(const v2f*)&lsH[buf][ko];
        a0 = a0 * sc + sh;
        a1 = a1 * sc + sh;
        a0.x = a0.x > 0.f ? a0.x : LEAK * a0.x;
        a0.y = a0.y > 0.f ? a0.y : LEAK * a0.y;
        a1.x = a1.x > 0.f ? a1.x : LEAK * a1.x;
        a1.y = a1.y > 0.f ? a1.y : LEAK * a1.y;
      }
      v2f b0 = *(const v2f*)&lsB[buf][c0][ko];
      v2f b1 = *(const v2f*)&lsB[buf][c0 + 16][ko];
      acc00 = wmma4(a0, b0, acc00);
      acc01 = wmma4(a0, b1, acc01);
      acc10 = wmma4(a1, b0, acc10);
      acc11 = wmma4(a1, b1, acc11);
    }
  };

  // ---- double-buffered pipeline: issue s+1, retire s (ASYNCcnt<=3), compute s
  const int nstages = K / BK;
  issue_stage(0, 0);
  int s = 0;
  for (; s < nstages - 1; ++s) {
    issue_stage((s + 1) & 1, (s + 1) * BK);
    async_wait_3();
    __syncthreads();
    compute_stage(s & 1);
    __syncthreads();
  }
  async_wait_0();
  __syncthreads();
  compute_stage(s & 1);

  // ---- epilogue: C/D layout VGPR r -> row (r, r+8), col = lane&15 ----
  #pragma unroll
  for (int sm = 0; sm < 2; ++sm) {
    #pragma unroll
    for (int sn = 0; sn < 2; ++sn) {
      v8f acc = (sm == 0) ? ((sn == 0) ? acc00 : acc01)
                          : ((sn == 0) ? acc10 : acc11);
      int cn = gn0 + wn * 32 + sn * 16 + l15;
      if (cn < N) {
        #pragma unroll
        for (int r = 0; r < 8; ++r) {
          int   row = gm0 + wm * 32 + sm * 16 + r + 8 * hi;
          float v   = acc[r];
          if (BIAS) v += bias[colC0 + cn];
          size_t off = (size_t)row * ldC + colC0 + cn;
          if (RESID) v += Cout[off];
          Cout[off] = v;
        }
      }
    }
  }
}

// ---------------------------------------------------------------------------
// Training-mode BatchNorm statistics over batch axis -> fused scale/shift
// ---------------------------------------------------------------------------
__global__ void bn_stats_kernel(const float* __restrict__ X,
                                const float* __restrict__ gamma,
                                const float* __restrict__ beta,
                                float* __restrict__ scale, float* __restrict__ shift,
                                int rows, int cols)
{
  int c = blockIdx.x * blockDim.x + threadIdx.x;
  if (c >= cols) return;
  float s = 0.f, s2 = 0.f;
  for (int r = 0; r < rows; ++r) {
    float v = X[(size_t)r * cols + c];
    s += v; s2 += v * v;
  }
  float m   = s / rows;
  float var = s2 / rows - m * m;
  float sc  = gamma[c] * rsqrtf(var + EPSN);
  scale[c]  = sc;
  shift[c]  = beta[c] - m * sc;
}

// codebook row squared-norms
__global__ void cnorm_kernel(const float* __restrict__ cb, float* __restrict__ out, int D)
{
  __shared__ float red[256];
  int n = blockIdx.x;
  float s = 0.f;
  for (int c = threadIdx.x; c < D; c += 256) {
    float v = cb[(size_t)n * D + c];
    s += v * v;
  }
  red[threadIdx.x] = s; __syncthreads();
  for (int k = 128; k > 0; k >>= 1) {
    if (threadIdx.x < k) red[threadIdx.x] += red[threadIdx.x + k];
    __syncthreads();
  }
  if (threadIdx.x == 0) out[n] = red[0];
}

// per-row argmin of (||c||^2 - 2 h.c)  (||h||^2 constant per row)
__global__ void argmin_kernel(const float* __restrict__ dot, const float* __restrict__ cn2,
                              int* __restrict__ idx, int rows, int nq)
{
  int b = blockIdx.x * blockDim.x + threadIdx.x;
  if (b >= rows) return;
  float best = 3.4e38f;
  int   bi   = 0;
  const float* dr = dot + (size_t)b * nq;
  for (int n = 0; n < nq; ++n) {
    float d = cn2[n] - 2.f * dr[n];
    if (d < best) { best = d; bi = n; }
  }
  idx[b] = bi;
}

// straight-through quantize in place + MSE accumulation
__global__ void quantize_kernel(float* __restrict__ h, const float* __restrict__ cb,
                                const int* __restrict__ idx, float* __restrict__ diff, int D)
{
  __shared__ float red[256];
  int b    = blockIdx.x;
  int code = idx[b];
  const float* q  = cb + (size_t)code * D;
  float*       hr = h  + (size_t)b * D;
  float s = 0.f;
  for (int c = threadIdx.x; c < D; c += 256) {
    float qa = q[c];
    float d  = qa - hr[c];
    s += d * d;
    hr[c] = qa;
  }
  red[threadIdx.x] = s; __syncthreads();
  for (int k = 128; k > 0; k >>= 1) {
    if (threadIdx.x < k) red[threadIdx.x] += red[threadIdx.x + k];
    __syncthreads();
  }
  if (threadIdx.x == 0) atomicAdd(diff, red[0]);
}

__global__ void set_scalar_kernel(float* p, float v) {
  if (threadIdx.x == 0 && blockIdx.x == 0) *p = v;
}
__global__ void mul_scalar_kernel(float* p, float v) {
  if (threadIdx.x == 0 && blockIdx.x == 0) *p *= v;
}

// ---------------------------------------------------------------------------
extern "C" void kernel_launch(void* const* d_in, const int* in_sizes, int n_in,
                              void* d_out, int out_size, void* d_ws, size_t ws_size,
                              hipStream_t stream)
{
  (void)in_sizes; (void)n_in; (void)out_size; (void)ws_size;

  // setup_inputs() dict order, then blocks[k] = {g0,b0,W0,bb0,g1,b1,W1,bb1,Ws}
  const float* x        = (const float*)d_in[0];
  const float* bn0_g    = (const float*)d_in[1];
  const float* bn0_b    = (const float*)d_in[2];
  const float* Wc       = (const float*)d_in[3];
  const float* bc       = (const float*)d_in[4];
  const float* codebook = (const float*)d_in[5];
  const float* ln_g     = (const float*)d_in[6];
  const float* ln_b     = (const float*)d_in[7];
  const float* Wout     = (const float*)d_in[8];
  const float* bout     = (const float*)d_in[9];
  const float* const* blk = (const float* const*)(d_in + 10);  // 9 per block

  // workspace layout (floats)
  float* ws = (float*)d_ws;
  const size_t N_HA   = (size_t)BATCH * 4096;   // h (quantized in place)
  const size_t N_DIST = (size_t)BATCH * 512;
  const size_t N_NET  = (size_t)BATCH * 2048;
  float* hA      = ws;
  float* dist    = hA   + N_HA;
  float* netbuf  = dist + N_DIST;
  float* hB      = netbuf + N_NET;
  float* scaleB  = hB   + N_NET;     // 4096
  float* shiftB  = scaleB + 4096;    // 4096
  float* cnormB  = shiftB + 4096;    // 512

  // outputs: ret[8192*256] floats, diff[1], indices[8192] (int bits)
  const size_t RET_ELEMS = (size_t)BATCH * 256;
  float* ret   = (float*)d_out;
  float* diffp = ret + RET_ELEMS;
  int*   idxp  = (int*)d_out + RET_ELEMS + 1;

  const int MT = BATCH / BM;  // 64 macro-tiles along M

  // --- bn0 + lrelu fused into fc_c A-fragments; fc_c GEMM 8192x1024 -> 8192x4096
  bn_stats_kernel<<<dim3(4), 256, 0, stream>>>(x, bn0_g, bn0_b, scaleB, shiftB, BATCH, 1024);
  gemm_wmma_f32<true, true, false><<<dim3(MT * (4096 / BN), 1), 256, 0, stream>>>(
      x, Wc, hA, scaleB, shiftB, bc, BATCH, 4096, 1024, 1024, 4096);

  // --- VQ: dot GEMM, argmin, straight-through quantize + diff
  cnorm_kernel<<<dim3(512), 256, 0, stream>>>(codebook, cnormB, 4096);
  gemm_wmma_f32<false, false, false><<<dim3(MT * (512 / BN), 1), 256, 0, stream>>>(
      hA, codebook, dist, nullptr, nullptr, nullptr, BATCH, 512, 4096, 4096, 512);
  argmin_kernel<<<dim3(BATCH / 256), 256, 0, stream>>>(dist, cnormB, idxp, BATCH, 512);
  set_scalar_kernel<<<1, 1, 0, stream>>>(diffp, 0.f);
  quantize_kernel<<<dim3(BATCH), 256, 0, stream>>>(hA, codebook, idxp, diffp, 4096);
  mul_scalar_kernel<<<1, 1, 0, stream>>>(diffp, 1.f / ((float)BATCH * 4096.f));

  // --- 3 grouped resnet blocks (G=8); buffer ping-pong hA -> hB -> hA -> hB
  float* bufIn[3]  = { hA, hB, hA };
  float* bufOut[3] = { hB, hA, hB };
  for (int k = 0; k < 3; ++k) {
    const int ci = 512 >> k, co = ci >> 1;
    const int Cin = 8 * ci, Co8 = 8 * co;
    const float* g0  = blk[9*k + 0]; const float* b0  = blk[9*k + 1];
    const float* W0  = blk[9*k + 2]; const float* bb0 = blk[9*k + 3];
    const float* g1  = blk[9*k + 4]; const float* b1  = blk[9*k + 5];
    const float* W1  = blk[9*k + 6]; const float* bb1 = blk[9*k + 7];
    const float* Ws  = blk[9*k + 8];
    const int nt = (co + BN - 1) / BN;

    bn_stats_kernel<<<dim3(Cin / 256), 256, 0, stream>>>(
        bufIn[k], g0, b0, scaleB, shiftB, BATCH, Cin);
    gemm_wmma_f32<true, true, false><<<dim3(MT * nt, 8), 256, 0, stream>>>(
        bufIn[k], W0, netbuf, scaleB, shiftB, bb0, BATCH, co, ci, Cin, Co8);
    bn_stats_kernel<<<dim3(Co8 / 256), 256, 0, stream>>>(
        netbuf, g1, b1, scaleB, shiftB, BATCH, Co8);
    gemm_wmma_f32<true, true, false><<<dim3(MT * nt, 8), 256, 0, stream>>>(
        netbuf, W1, bufOut[k], scaleB, shiftB, bb1, BATCH, co, co, Co8, Co8);
    // shortcut: h_next = gconv(h, Ws) + dx   (raw A, no bias, residual add)
    gemm_wmma_f32<false, false, true><<<dim3(MT * nt, 8), 256, 0, stream>>>(
        bufIn[k], Ws, bufOut[k], nullptr, nullptr, nullptr, BATCH, co, ci, Cin, Co8);
  }

  // --- final: lrelu(bnorm(h)) -> grouped Wout [8,32,64] -> ret [8192,8,32]
  bn_stats_kernel<<<dim3(2), 256, 0, stream>>>(hB, ln_g, ln_b, scaleB, shiftB, BATCH, 512);
  gemm_wmma_f32<true, true, false><<<dim3(MT * 1, 8), 256, 0, stream>>>(
      hB, Wout, ret, scaleB, shiftB, bout, BATCH, 32, 64, 512, 256);
}